// GAT_38465727103648
// MI455X (gfx1250) — compile-verified
//
#include <hip/hip_runtime.h>
#include <hip/hip_bf16.h>

typedef __attribute__((ext_vector_type(2))) float v2f;
typedef __attribute__((ext_vector_type(8))) float v8f;

#define NEG_SLOPE 0.2f

// ---------------- utility fills ----------------
__global__ void fill_f32(float* p, float v, int n) {
    int i = blockIdx.x * blockDim.x + threadIdx.x;
    if (i < n) p[i] = v;
}
__global__ void fill_i32(int* p, int v, int n) {
    int i = blockIdx.x * blockDim.x + threadIdx.x;
    if (i < n) p[i] = v;
}

// order-preserving float<->int for atomicMax on floats
__device__ __forceinline__ int fenc(float f) {
    int i = __float_as_int(f);
    return (i >= 0) ? i : (i ^ 0x7FFFFFFF);
}
__device__ __forceinline__ float fdec(int k) {
    return __int_as_float((k >= 0) ? k : (k ^ 0x7FFFFFFF));
}
__device__ __forceinline__ float lrelu(float x) {
    return x >= 0.0f ? x : NEG_SLOPE * x;
}

// ---------------- B pre-pack into WMMA lane order ----------------
// For col-tile t, k-quad kq, lane l: lane holds B[kq*4 + 2*(l>>4) + {0,1}][t*16 + (l&15)]
// stored contiguously: Bp[((t*(K/4) + kq)*32 + l)*2 + {0,1}].  Cols >= Nc are zero-padded.
__global__ void pack_B(const float* __restrict__ B, float* __restrict__ Bp,
                       int K, int Nc, int Tc)
{
    int i = blockIdx.x * blockDim.x + threadIdx.x;
    int total = Tc * (K / 4) * 32;
    if (i >= total) return;
    int lane = i & 31;
    int rest = i >> 5;
    int kq   = rest % (K / 4);
    int t    = rest / (K / 4);
    int col  = t * 16 + (lane & 15);
    int k    = kq * 4 + (lane >> 4) * 2;
    bool ok  = col < Nc;
    Bp[(size_t)i * 2 + 0] = ok ? B[(size_t)k * Nc + col]       : 0.0f;
    Bp[(size_t)i * 2 + 1] = ok ? B[(size_t)(k + 1) * Nc + col] : 0.0f;
}

// ---------------- WMMA fp32 GEMM with packed B ----------------
// One wave per 16x16 tile; K is compile-time -> fully unrolled v_wmma chain.
// Inner body: one global_load_b64 (A pair) + one global_load_b64 (packed B pair) + v_wmma.
template <int K>
__global__ void __launch_bounds__(32) wmma_gemm_f32(
    const float* __restrict__ A, const float* __restrict__ Bp,
    float* __restrict__ C, int Nc)
{
    const int m0   = blockIdx.x * 16;
    const int t    = blockIdx.y;       // column tile
    const int lane = threadIdx.x;
    const int half = lane >> 4;
    const int lm   = lane & 15;

    const float* ap = A  + (size_t)(m0 + lm) * K + half * 2;
    const float* bp = Bp + ((size_t)t * (K / 4)) * 64 + lane * 2;

    v8f acc = {};
    #pragma unroll
    for (int k = 0; k < K / 4; ++k) {
        v2f a = *(const v2f*)ap;  ap += 4;
        v2f b = *(const v2f*)bp;  bp += 64;
        acc = __builtin_amdgcn_wmma_f32_16x16x4_f32(
            false, a, false, b, (short)0, acc, false, false);
    }

    const int col = t * 16 + lm;
    if (col < Nc) {
        #pragma unroll
        for (int v = 0; v < 8; ++v) {
            const int row = m0 + v + 8 * half;   // C/D: VGPR v -> row v + 8*(lane/16)
            C[(size_t)row * Nc + col] = acc[v];
        }
    }
}

// ---------------- layer-1 attention coefficients per (node, head) ----------------
__global__ void alpha1_kernel(const float* __restrict__ hW,
                              const float* __restrict__ a_src,
                              const float* __restrict__ a_dst,
                              float* __restrict__ as1, float* __restrict__ ad1, int N)
{
    int i = blockIdx.x * blockDim.x + threadIdx.x;   // i = n*8 + h
    if (i >= N * 8) return;
    int n = i >> 3, h = i & 7;
    const float* row = hW + (size_t)n * 64 + h * 8;
    float s = 0.0f, d = 0.0f;
    #pragma unroll
    for (int c = 0; c < 8; ++c) {
        s += row[c] * a_src[h * 8 + c];
        d += row[c] * a_dst[h * 8 + c];
    }
    as1[i] = s;
    ad1[i] = d;
}

// ---------------- edge helpers ----------------
__device__ __forceinline__ void edge_ids(const long long* ei, long long e,
                                         int E, int* src, int* dst)
{
    if (e < E) { *src = (int)ei[e]; *dst = (int)ei[(long long)E + e]; }
    else       { int n = (int)(e - E); *src = n; *dst = n; }  // self loop
}

// ---------------- layer-1 edge passes ----------------
// max/sum passes: one thread per edge, vectorized 32B gathers of the 8-head alphas
__global__ void l1_pass_max(const long long* __restrict__ ei, int E, int E2,
                            const float* __restrict__ as, const float* __restrict__ ad,
                            int* __restrict__ m)
{
    long long e = (long long)blockIdx.x * blockDim.x + threadIdx.x;
    if (e >= E2) return;
    int src, dst; edge_ids(ei, e, E, &src, &dst);
    const float4* s4 = (const float4*)(as + (size_t)src * 8);
    const float4* d4 = (const float4*)(ad + (size_t)dst * 8);
    float4 s0 = s4[0], s1 = s4[1], d0 = d4[0], d1 = d4[1];
    float sv[8] = {s0.x, s0.y, s0.z, s0.w, s1.x, s1.y, s1.z, s1.w};
    float dv[8] = {d0.x, d0.y, d0.z, d0.w, d1.x, d1.y, d1.z, d1.w};
    int* mrow = m + (size_t)dst * 8;
    #pragma unroll
    for (int h = 0; h < 8; ++h)
        atomicMax(&mrow[h], fenc(lrelu(sv[h] + dv[h])));
}

__global__ void l1_pass_sum(const long long* __restrict__ ei, int E, int E2,
                            const float* __restrict__ as, const float* __restrict__ ad,
                            const int* __restrict__ m, float* __restrict__ z)
{
    long long e = (long long)blockIdx.x * blockDim.x + threadIdx.x;
    if (e >= E2) return;
    int src, dst; edge_ids(ei, e, E, &src, &dst);
    const float4* s4 = (const float4*)(as + (size_t)src * 8);
    const float4* d4 = (const float4*)(ad + (size_t)dst * 8);
    float4 s0 = s4[0], s1 = s4[1], d0 = d4[0], d1 = d4[1];
    float sv[8] = {s0.x, s0.y, s0.z, s0.w, s1.x, s1.y, s1.z, s1.w};
    float dv[8] = {d0.x, d0.y, d0.z, d0.w, d1.x, d1.y, d1.z, d1.w};
    const int* mrow = m + (size_t)dst * 8;
    float* zrow = z + (size_t)dst * 8;
    #pragma unroll
    for (int h = 0; h < 8; ++h) {
        float p = expf(lrelu(sv[h] + dv[h]) - fdec(mrow[h]));
        atomicAdd(&zrow[h], p);
    }
}

// aggregation: one thread per (edge, head) — dominated by its 8 f32 atomics
__global__ void l1_pass_agg(const long long* __restrict__ ei, int E, int E2,
                            const float* __restrict__ as, const float* __restrict__ ad,
                            const int* __restrict__ m, const float* __restrict__ z,
                            const float* __restrict__ hW, float* __restrict__ out1)
{
    long long t = (long long)blockIdx.x * blockDim.x + threadIdx.x;
    if (t >= (long long)E2 * 8) return;
    int h = (int)(t & 7); long long e = t >> 3;
    int src, dst; edge_ids(ei, e, E, &src, &dst);
    float ev   = lrelu(as[src * 8 + h] + ad[dst * 8 + h]);
    float p    = expf(ev - fdec(m[dst * 8 + h]));
    float coef = p / (z[dst * 8 + h] + 1e-16f);
    const float* hrow = hW + (size_t)src * 64 + h * 8;
    float* orow = out1 + (size_t)dst * 64 + h * 8;
    #pragma unroll
    for (int c = 0; c < 8; ++c) atomicAdd(&orow[c], coef * hrow[c]);
}

// ---------------- bias + ReLU (in place) ----------------
__global__ void bias_relu_kernel(float* __restrict__ out1,
                                 const float* __restrict__ b1, int N)
{
    int i = blockIdx.x * blockDim.x + threadIdx.x;
    if (i >= N * 64) return;
    out1[i] = fmaxf(out1[i] + b1[i & 63], 0.0f);
}

// ---------------- layer-2 attention coefficients per node (H=1) ----------------
__global__ void alpha2_kernel(const float* __restrict__ h2,
                              const float* __restrict__ a_src,
                              const float* __restrict__ a_dst,
                              float* __restrict__ as2, float* __restrict__ ad2, int N)
{
    int n = blockIdx.x * blockDim.x + threadIdx.x;
    if (n >= N) return;
    const float* row = h2 + (size_t)n * 8;
    float s = 0.0f, d = 0.0f;
    #pragma unroll
    for (int c = 0; c < 8; ++c) { s += row[c] * a_src[c]; d += row[c] * a_dst[c]; }
    as2[n] = s;
    ad2[n] = d;
}

// ---------------- layer-2 edge passes (thread per edge, H=1) ----------------
__global__ void l2_pass_max(const long long* __restrict__ ei, int E, int E2,
                            const float* __restrict__ as, const float* __restrict__ ad,
                            int* __restrict__ m)
{
    long long e = (long long)blockIdx.x * blockDim.x + threadIdx.x;
    if (e >= E2) return;
    int src, dst; edge_ids(ei, e, E, &src, &dst);
    atomicMax(&m[dst], fenc(lrelu(as[src] + ad[dst])));
}

__global__ void l2_pass_sum(const long long* __restrict__ ei, int E, int E2,
                            const float* __restrict__ as, const float* __restrict__ ad,
                            const int* __restrict__ m, float* __restrict__ z)
{
    long long e = (long long)blockIdx.x * blockDim.x + threadIdx.x;
    if (e >= E2) return;
    int src, dst; edge_ids(ei, e, E, &src, &dst);
    float p = expf(lrelu(as[src] + ad[dst]) - fdec(m[dst]));
    atomicAdd(&z[dst], p);
}

__global__ void l2_pass_agg(const long long* __restrict__ ei, int E, int E2,
                            const float* __restrict__ as, const float* __restrict__ ad,
                            const int* __restrict__ m, const float* __restrict__ z,
                            const float* __restrict__ h2, float* __restrict__ out2)
{
    long long e = (long long)blockIdx.x * blockDim.x + threadIdx.x;
    if (e >= E2) return;
    int src, dst; edge_ids(ei, e, E, &src, &dst);
    float p    = expf(lrelu(as[src] + ad[dst]) - fdec(m[dst]));
    float coef = p / (z[dst] + 1e-16f);
    const float* hrow = h2 + (size_t)src * 8;
    float* orow = out2 + (size_t)dst * 8;
    #pragma unroll
    for (int c = 0; c < 8; ++c) atomicAdd(&orow[c], coef * hrow[c]);
}

// ---------------- epilogue: embeddings + classifier ----------------
__global__ void final_kernel(const float* __restrict__ out2,
                             const float* __restrict__ b2,
                             const float* __restrict__ Wc,
                             const float* __restrict__ bc,
                             float* __restrict__ d_out, int N)
{
    int n = blockIdx.x * blockDim.x + threadIdx.x;
    if (n >= N) return;
    float emb[8];
    float l0 = bc[0], l1 = bc[1];
    #pragma unroll
    for (int c = 0; c < 8; ++c) {
        emb[c] = out2[(size_t)n * 8 + c] + b2[c];
        d_out[(size_t)n * 8 + c] = emb[c];
        l0 += emb[c] * Wc[c * 2 + 0];
        l1 += emb[c] * Wc[c * 2 + 1];
    }
    float* logits = d_out + (size_t)N * 8;
    logits[(size_t)n * 2 + 0] = l0;
    logits[(size_t)n * 2 + 1] = l1;
}

// ---------------- host orchestration ----------------
extern "C" void kernel_launch(void* const* d_in, const int* in_sizes, int n_in,
                              void* d_out, int out_size, void* d_ws, size_t ws_size,
                              hipStream_t stream) {
    const float*     x      = (const float*)d_in[0];
    const long long* ei     = (const long long*)d_in[1];
    const float*     W1     = (const float*)d_in[2];
    const float*     a_src1 = (const float*)d_in[3];
    const float*     a_dst1 = (const float*)d_in[4];
    const float*     b1     = (const float*)d_in[5];
    const float*     W2     = (const float*)d_in[6];
    const float*     a_src2 = (const float*)d_in[7];
    const float*     a_dst2 = (const float*)d_in[8];
    const float*     b2     = (const float*)d_in[9];
    const float*     Wc     = (const float*)d_in[10];
    const float*     bc     = (const float*)d_in[11];

    const int N  = in_sizes[0] / 128;   // 100000
    const int E  = in_sizes[1] / 2;     // 3200000
    const int E2 = E + N;               // with self loops

    // workspace layout
    float* hW   = (float*)d_ws;                 // [N,64] pre-agg features
    float* out1 = hW   + (size_t)64 * N;        // [N,64] aggregated (then h1)
    float* as1  = out1 + (size_t)64 * N;        // [N,8]
    float* ad1  = as1  + (size_t)8  * N;        // [N,8]
    int*   m1   = (int*)(ad1 + (size_t)8 * N);  // [N,8] encoded max
    float* z1   = (float*)(m1 + (size_t)8 * N); // [N,8]
    float* h2   = z1   + (size_t)8 * N;         // [N,8]
    float* as2  = h2   + (size_t)8 * N;         // [N]
    float* ad2  = as2  + (size_t)N;             // [N]
    int*   m2   = (int*)(ad2 + (size_t)N);      // [N]
    float* z2   = (float*)(m2 + (size_t)N);     // [N]
    float* out2 = z2   + (size_t)N;             // [N,8]
    float* Bp1  = out2 + (size_t)8 * N;         // [4*32*32*2]  = 8192  (W1 packed)
    float* Bp2  = Bp1  + 8192;                  // [1*16*32*2]  = 1024  (W2 packed, padded)

    const int B = 256;
    #define NB(n) ((int)(((long long)(n) + B - 1) / B))

    // init accumulators (every call; graph-replay safe)
    fill_f32<<<NB(64 * N), B, 0, stream>>>(out1, 0.0f, 64 * N);
    fill_f32<<<NB(8 * N),  B, 0, stream>>>(z1,   0.0f, 8 * N);
    fill_i32<<<NB(8 * N),  B, 0, stream>>>(m1, (int)0x80000000, 8 * N);
    fill_f32<<<NB(8 * N),  B, 0, stream>>>(out2, 0.0f, 8 * N);
    fill_f32<<<NB(N),      B, 0, stream>>>(z2,   0.0f, N);
    fill_i32<<<NB(N),      B, 0, stream>>>(m2, (int)0x80000000, N);

    // pack weights into WMMA lane order (zero-pad W2's 8 cols to a 16-wide tile)
    pack_B<<<NB(4 * 32 * 32), B, 0, stream>>>(W1, Bp1, 128, 64, 4);
    pack_B<<<NB(1 * 16 * 32), B, 0, stream>>>(W2, Bp2,  64,  8, 1);

    // layer 1: hW = x @ W1   (M=100000, K=128, Nc=64)  -> 32 v_wmma per wave
    wmma_gemm_f32<128><<<dim3(N / 16, 4), 32, 0, stream>>>(x, Bp1, hW, 64);
    alpha1_kernel<<<NB(8 * N), B, 0, stream>>>(hW, a_src1, a_dst1, as1, ad1, N);

    l1_pass_max<<<NB(E2), B, 0, stream>>>(ei, E, E2, as1, ad1, m1);
    l1_pass_sum<<<NB(E2), B, 0, stream>>>(ei, E, E2, as1, ad1, m1, z1);
    const long long T1 = (long long)E2 * 8;
    l1_pass_agg<<<NB(T1), B, 0, stream>>>(ei, E, E2, as1, ad1, m1, z1, hW, out1);
    bias_relu_kernel<<<NB(64 * N), B, 0, stream>>>(out1, b1, N);

    // layer 2: h2 = h1 @ W2  (M=100000, K=64, Nc=8)  -> 16 v_wmma per wave
    wmma_gemm_f32<64><<<dim3(N / 16, 1), 32, 0, stream>>>(out1, Bp2, h2, 8);
    alpha2_kernel<<<NB(N), B, 0, stream>>>(h2, a_src2, a_dst2, as2, ad2, N);

    l2_pass_max<<<NB(E2), B, 0, stream>>>(ei, E, E2, as2, ad2, m2);
    l2_pass_sum<<<NB(E2), B, 0, stream>>>(ei, E, E2, as2, ad2, m2, z2);
    l2_pass_agg<<<NB(E2), B, 0, stream>>>(ei, E, E2, as2, ad2, m2, z2, h2, out2);

    // epilogue: embeddings + logits into d_out
    final_kernel<<<NB(N), B, 0, stream>>>(out2, b2, Wc, bc, (float*)d_out, N);
    #undef NB
}